// GAT_VAE_9981503996080
// MI455X (gfx1250) — compile-verified
//
#include <hip/hip_runtime.h>
#include <math.h>

#define NN 20000
#define NE 320000
#define HID 128
#define HID2 256

typedef float v2f __attribute__((ext_vector_type(2)));
typedef float v8f __attribute__((ext_vector_type(8)));

// ---------------------------------------------------------------------------
// GEMM: C[M x Fout] = A[M x K] * W[Fout x K]^T   (W row-major [Fout,K])
//
// Block = 256 threads (8 wave32). Grid.x = ntx * ceil(Mtiles/8).
// Each block stages W[col0 .. col0+63, 0:K] (32/64 KB) into LDS once, then
// each wave computes a 16x64 tile of C with V_WMMA_F32_16X16X4_F32,
// B fragments sourced from LDS (ds_load_b64), A fragments from global.
//
// A-tile (16x4 f32): lane = m + 16*(k/2), vgpr = k%2
// B-tile (4x16 f32): lane = n + 16*(k/2), vgpr = k%2
// C/D (16x16 f32):  vgpr r -> row (r + 8*(lane/16)), col = lane%16
// ---------------------------------------------------------------------------
__global__ void gemm_wmma_f32(const float* __restrict__ A,
                              const float* __restrict__ W,
                              float* __restrict__ C,
                              int M, int K, int ntx, int ldc, int col_off) {
    extern __shared__ float wlds[];            // 64 * K floats
    const int tid  = threadIdx.x;
    const int lane = tid & 31;
    const int wid  = tid >> 5;                 // 0..7
    const int mtiles = M >> 4;

    const int ct   = (int)blockIdx.x % ntx;
    const int grp  = (int)blockIdx.x / ntx;
    const int col0 = ct << 6;

    // cooperative stage of 64 contiguous weight rows into LDS (float4 copies)
    const float4* wsrc = (const float4*)(W + (size_t)col0 * K);
    float4*       wdst = (float4*)wlds;
    const int total4 = (64 * K) >> 2;
    for (int i = tid; i < total4; i += 256) {
        wdst[i] = wsrc[i];
    }
    __syncthreads();

    const int rt = grp * 8 + wid;
    if (rt >= mtiles) return;                  // wave-uniform: EXEC stays full
    const int row0 = rt << 4;
    const int ml   = lane & 15;
    const int half = lane >> 4;

    v8f acc[4] = {};

    const float* arow = A + (size_t)(row0 + ml) * K + 2 * half;
    const float* brow = wlds + (size_t)ml * K + 2 * half;

    for (int k0 = 0; k0 < K; k0 += 4) {
        v2f a = *(const v2f*)(arow + k0);
#pragma unroll
        for (int t = 0; t < 4; ++t) {
            v2f b = *(const v2f*)(brow + (size_t)t * 16 * K + k0);
            acc[t] = __builtin_amdgcn_wmma_f32_16x16x4_f32(
                false, a, false, b, (short)0, acc[t], false, false);
        }
    }

#pragma unroll
    for (int t = 0; t < 4; ++t) {
        const int n = col_off + col0 + t * 16 + ml;
#pragma unroll
        for (int r = 0; r < 8; ++r) {
            C[(size_t)(row0 + r + 8 * half) * ldc + n] = acc[t][r];
        }
    }
}

// ---------------------------------------------------------------------------
// Per-node attention scalars: p[n] = z[n]·w_att[0:F], q[n] = z[n]·w_att[F:2F]
// One wave per node; wave32 shuffle reduction.
// ---------------------------------------------------------------------------
__global__ void node_pq(const float* __restrict__ z,
                        const float* __restrict__ watt,
                        float* __restrict__ p, float* __restrict__ q,
                        int n, int F) {
    const int wave = (int)((blockIdx.x * blockDim.x + threadIdx.x) >> 5);
    const int lane = threadIdx.x & 31;
    if (wave >= n) return;
    const float* zr = z + (size_t)wave * F;
    float sp = 0.f, sq = 0.f;
    for (int f = lane; f < F; f += 32) {
        float zv = zr[f];
        sp += zv * watt[f];
        sq += zv * watt[F + f];
    }
#pragma unroll
    for (int o = 16; o > 0; o >>= 1) {
        sp += __shfl_xor(sp, o, 32);
        sq += __shfl_xor(sq, o, 32);
    }
    if (lane == 0) { p[wave] = sp; q[wave] = sq; }
}

// init m=-inf, denom=0, agg=0
__global__ void seg_init(float* __restrict__ m, float* __restrict__ den,
                         float* __restrict__ agg, int n, int logF) {
    const int total = n << logF;
    for (int i = (int)(blockIdx.x * blockDim.x + threadIdx.x); i < total;
         i += (int)(gridDim.x * blockDim.x)) {
        agg[i] = 0.f;
        if (i < n) { m[i] = -INFINITY; den[i] = 0.f; }
    }
}

__device__ __forceinline__ float leaky(float e) {
    return e > 0.f ? e : 0.01f * e;
}

// sign-aware float atomic max (monotonic int/uint trick; init must be -inf)
__device__ __forceinline__ void atomicMaxF(float* addr, float v) {
    if (v >= 0.f) atomicMax((int*)addr, __float_as_int(v));
    else          atomicMin((unsigned int*)addr, __float_as_uint(v));
}

__global__ void edge_max(const int* __restrict__ src, const int* __restrict__ dst,
                         const float* __restrict__ p, const float* __restrict__ q,
                         float* __restrict__ m, int E) {
    int i = (int)(blockIdx.x * blockDim.x + threadIdx.x);
    if (i >= E) return;
    int s = src[i], d = dst[i];
    atomicMaxF(&m[d], leaky(p[s] + q[d]));
}

__global__ void edge_sum(const int* __restrict__ src, const int* __restrict__ dst,
                         const float* __restrict__ p, const float* __restrict__ q,
                         const float* __restrict__ m, float* __restrict__ den, int E) {
    int i = (int)(blockIdx.x * blockDim.x + threadIdx.x);
    if (i >= E) return;
    int s = src[i], d = dst[i];
    float e = leaky(p[s] + q[d]);
    atomicAdd(&den[d], __expf(e - m[d]));
}

// one wave per edge: agg[dst] += coef * z[src]
__global__ void edge_agg(const int* __restrict__ src, const int* __restrict__ dst,
                         const float* __restrict__ p, const float* __restrict__ q,
                         const float* __restrict__ m, const float* __restrict__ den,
                         const float* __restrict__ z, float* __restrict__ agg,
                         int E, int F) {
    const int wave = (int)((blockIdx.x * blockDim.x + threadIdx.x) >> 5);
    const int lane = threadIdx.x & 31;
    if (wave >= E) return;
    int s = src[wave], d = dst[wave];
    float e = leaky(p[s] + q[d]);
    float coef = __expf(e - m[d]) / den[d];
    const int nch = F >> 5;                      // 4 (F=128) or 8 (F=256)
    const float* zr = z + (size_t)s * F + lane * nch;
    float*       ar = agg + (size_t)d * F + lane * nch;
    for (int j = 0; j < nch; j += 4) {
        float4 zv = *(const float4*)(zr + j);
        atomicAdd(ar + j + 0, coef * zv.x);
        atomicAdd(ar + j + 1, coef * zv.y);
        atomicAdd(ar + j + 2, coef * zv.z);
        atomicAdd(ar + j + 3, coef * zv.w);
    }
}

// out[n, off+f] = h_in[n, f] + relu(hs[n,f] + agg[n,f])
__global__ void finalize(const float* __restrict__ h_in, int ld_in,
                         const float* __restrict__ hs, const float* __restrict__ agg,
                         float* __restrict__ out, int ld_out, int col_off,
                         int n, int logF) {
    const int total = n << logF;
    int i = (int)(blockIdx.x * blockDim.x + threadIdx.x);
    if (i >= total) return;
    int node = i >> logF;
    int f = i & ((1 << logF) - 1);
    float v = hs[i] + agg[i];
    v = v > 0.f ? v : 0.f;
    out[(size_t)node * ld_out + col_off + f] = h_in[(size_t)node * ld_in + f] + v;
}

// ---------------------------------------------------------------------------
// Host-side: one GAT layer (single head)
// ---------------------------------------------------------------------------
static void run_gat_layer(const float* h_in, int F,
                          const float* w_self, const float* w_func, const float* w_att,
                          const int* src, const int* dst,
                          float* out, int ld_out, int col_off,
                          float* zbuf, float* hsbuf, float* aggbuf,
                          float* pbuf, float* qbuf, float* mbuf, float* dbuf,
                          hipStream_t stream) {
    const int logF = (F == 128) ? 7 : 8;
    const int ntx = F >> 6;
    const int mtiles = NN >> 4;                      // 1250
    const int gemm_blocks = ntx * ((mtiles + 7) / 8);
    const size_t lds_bytes = (size_t)64 * F * sizeof(float);  // 32KB / 64KB

    gemm_wmma_f32<<<gemm_blocks, 256, lds_bytes, stream>>>(h_in, w_self, hsbuf, NN, F, ntx, F, 0);
    gemm_wmma_f32<<<gemm_blocks, 256, lds_bytes, stream>>>(h_in, w_func, zbuf,  NN, F, ntx, F, 0);

    node_pq<<<(NN * 32 + 255) / 256, 256, 0, stream>>>(zbuf, w_att, pbuf, qbuf, NN, F);

    seg_init<<<2048, 256, 0, stream>>>(mbuf, dbuf, aggbuf, NN, logF);

    edge_max<<<(NE + 255) / 256, 256, 0, stream>>>(src, dst, pbuf, qbuf, mbuf, NE);
    edge_sum<<<(NE + 255) / 256, 256, 0, stream>>>(src, dst, pbuf, qbuf, mbuf, dbuf, NE);
    edge_agg<<<(NE * 32 + 255) / 256, 256, 0, stream>>>(src, dst, pbuf, qbuf, mbuf, dbuf,
                                                        zbuf, aggbuf, NE, F);

    const int total = NN << logF;
    finalize<<<(total + 255) / 256, 256, 0, stream>>>(h_in, F, hsbuf, aggbuf,
                                                      out, ld_out, col_off, NN, logF);
}

extern "C" void kernel_launch(void* const* d_in, const int* in_sizes, int n_in,
                              void* d_out, int out_size, void* d_ws, size_t ws_size,
                              hipStream_t stream) {
    (void)in_sizes; (void)n_in; (void)out_size; (void)ws_size;

    const float* h         = (const float*)d_in[0];
    // d_in[1] e_w, d_in[2] snorm_n: dead w.r.t. output (embedding_e unused)
    const int*   edge_src  = (const int*)d_in[3];
    const int*   edge_dst  = (const int*)d_in[4];
    const float* w_self_1a = (const float*)d_in[5];
    const float* w_func_1a = (const float*)d_in[6];
    const float* w_att_1a  = (const float*)d_in[7];
    const float* w_self_1b = (const float*)d_in[8];
    const float* w_func_1b = (const float*)d_in[9];
    const float* w_att_1b  = (const float*)d_in[10];
    const float* w_self_2  = (const float*)d_in[13];
    const float* w_func_2  = (const float*)d_in[14];
    const float* w_att_2   = (const float*)d_in[15];

    // workspace carve-up (floats): h1 | z | hs | agg | p | q | m | denom
    float* ws   = (float*)d_ws;
    float* h1   = ws;                       // NN*256
    float* zbuf = h1   + (size_t)NN * 256;  // NN*256
    float* hsbf = zbuf + (size_t)NN * 256;  // NN*256
    float* agg  = hsbf + (size_t)NN * 256;  // NN*256
    float* pbuf = agg  + (size_t)NN * 256;  // NN
    float* qbuf = pbuf + NN;                // NN
    float* mbuf = qbuf + NN;                // NN
    float* dbuf = mbuf + NN;                // NN

    // ---- layer 1, head a -> h1[:, 0:128] ----
    run_gat_layer(h, HID, w_self_1a, w_func_1a, w_att_1a, edge_src, edge_dst,
                  h1, HID2, 0, zbuf, hsbf, agg, pbuf, qbuf, mbuf, dbuf, stream);
    // ---- layer 1, head b -> h1[:, 128:256] ----
    run_gat_layer(h, HID, w_self_1b, w_func_1b, w_att_1b, edge_src, edge_dst,
                  h1, HID2, HID, zbuf, hsbf, agg, pbuf, qbuf, mbuf, dbuf, stream);
    // ---- layer 2 -> d_out ----
    run_gat_layer(h1, HID2, w_self_2, w_func_2, w_att_2, edge_src, edge_dst,
                  (float*)d_out, HID2, 0, zbuf, hsbf, agg, pbuf, qbuf, mbuf, dbuf, stream);
}